// ResGCN_19009525252200
// MI455X (gfx1250) — compile-verified
//
#include <hip/hip_runtime.h>
#include <hip/hip_bf16.h>

typedef __attribute__((ext_vector_type(2))) float v2f;
typedef __attribute__((ext_vector_type(8))) float v8f;

#define H_FEATS 64
#define BN_EPS 1e-5f

// ---------------- utility kernels ----------------

__global__ __launch_bounds__(256)
void zero_kernel(float* __restrict__ p, long count) {
    long i = (long)blockIdx.x * blockDim.x + threadIdx.x;
    long stride = (long)gridDim.x * blockDim.x;
    for (; i < count; i += stride) p[i] = 0.0f;
}

__global__ __launch_bounds__(256)
void degree_kernel(const int* __restrict__ src, const int* __restrict__ dst,
                   float* __restrict__ deg_out, float* __restrict__ deg_in, int n_edges) {
    int e = blockIdx.x * blockDim.x + threadIdx.x;
    if (e < n_edges) {
        atomicAdd(&deg_out[src[e]], 1.0f);
        atomicAdd(&deg_in[dst[e]], 1.0f);
    }
}

__global__ __launch_bounds__(256)
void norm_kernel(float* __restrict__ deg, long count) {
    long i = (long)blockIdx.x * blockDim.x + threadIdx.x;
    if (i < count) {
        float d = deg[i];
        d = (d < 1.0f) ? 1.0f : d;
        deg[i] = rsqrtf(d);
    }
}

// one 32-lane wave per edge, F == 64: coalesced gather of y[src,:],
// coalesced float atomics into agg[dst,:]
__global__ __launch_bounds__(256)
void scatter_kernel(const float* __restrict__ y,
                    const int* __restrict__ src, const int* __restrict__ dst,
                    float* __restrict__ agg, int n_edges) {
    int wave = (int)(((long)blockIdx.x * blockDim.x + threadIdx.x) >> 5);
    int lane = threadIdx.x & 31;
    if (wave >= n_edges) return;
    int s = src[wave];
    int d = dst[wave];
    const float* ys = y + (long)s * H_FEATS;
    float* ad = agg + (long)d * H_FEATS;
    atomicAdd(&ad[lane], ys[lane]);
    atomicAdd(&ad[lane + 32], ys[lane + 32]);
}

// ---------------- fp32 WMMA GEMM ----------------
// C[n x nout] = (A[n x K] * rownorm[row]) @ W[K x nout] (+ bias)
// Block = 4 waves; each wave computes one 16-row x (NT*16)-col strip.
// W is staged in LDS (zero-padded cols), B fragments read via ds_load.
// A frag (16x4 f32, 2 VGPR/lane): lane M = lane&15, K = k + 2*(lane>>4) + {0,1}
// B frag (4x16, 2 VGPR/lane): lane N = lane&15, K rows mirrored like A.
template <int NT>
__global__ __launch_bounds__(128)
void gemm_wmma_f32(const float* __restrict__ A, const float* __restrict__ rownorm,
                   const float* __restrict__ W, const float* __restrict__ bias,
                   float* __restrict__ C, int n, int K, int nout) {
    extern __shared__ float sW[];            // K x (NT*16), zero-padded columns
    const int W16 = NT * 16;

    // cooperative staging of W into LDS
    for (int idx = threadIdx.x; idx < K * W16; idx += blockDim.x) {
        int k = idx / W16;
        int c = idx - k * W16;
        sW[idx] = (c < nout) ? W[(long)k * nout + c] : 0.0f;
    }
    __syncthreads();

    const int wave = threadIdx.x >> 5;
    const int lane = threadIdx.x & 31;
    const int half = lane >> 4;
    const int l    = lane & 15;
    const int kb   = 2 * half;

    const int m0 = (blockIdx.x * 4 + wave) * 16;
    if (m0 >= n) return;                     // whole-wave uniform exit

    int row = m0 + l;
    if (row >= n) row = n - 1;               // clamp (n % 16 == 0 in practice)
    const float rs = rownorm ? rownorm[row] : 1.0f;
    const float* arow = A + (long)row * K + kb;

    v8f acc[NT];
#pragma unroll
    for (int t = 0; t < NT; ++t)
        acc[t] = (v8f){0.f, 0.f, 0.f, 0.f, 0.f, 0.f, 0.f, 0.f};

    for (int k = 0; k < K; k += 4) {
        v2f a;
        a.x = arow[k]     * rs;
        a.y = arow[k + 1] * rs;
        const float* w0 = sW + (k + kb) * W16 + l;
        const float* w1 = w0 + W16;
#pragma unroll
        for (int t = 0; t < NT; ++t) {
            v2f b;
            b.x = w0[t * 16];
            b.y = w1[t * 16];
            acc[t] = __builtin_amdgcn_wmma_f32_16x16x4_f32(false, a, false, b,
                                                           (short)0, acc[t], false, false);
        }
    }

#pragma unroll
    for (int t = 0; t < NT; ++t) {
        int col = t * 16 + l;
        if (col < nout) {
            float bv = bias ? bias[col] : 0.0f;
#pragma unroll
            for (int v = 0; v < 8; ++v) {
                int r = m0 + v + 8 * half;   // C/D layout: VGPR v -> M = v (+8 upper half)
                if (r < n)
                    C[(long)r * nout + col] = acc[t][v] + bv;
            }
        }
    }
}

// ---------------- batch norm (folds z = agg*in_norm + bias on the fly) ----------------

__global__ __launch_bounds__(256)
void bn_stats_kernel(const float* __restrict__ agg, const float* __restrict__ innorm,
                     const float* __restrict__ bias,
                     float* __restrict__ sum, float* __restrict__ sumsq, int n) {
    __shared__ float s1[256];
    __shared__ float s2[256];
    int tid = threadIdx.x;
    int col = tid & 63;
    int gid = blockIdx.x * 256 + tid;
    int rstart  = gid >> 6;
    int rstride = (gridDim.x * 256) >> 6;
    float bv = bias[col];
    float a = 0.0f, b = 0.0f;
    for (int r = rstart; r < n; r += rstride) {
        float v = agg[(long)r * H_FEATS + col] * innorm[r] + bv;
        a += v;
        b += v * v;
    }
    s1[tid] = a;
    s2[tid] = b;
    __syncthreads();
    if (tid < 64) {
        a = s1[tid] + s1[tid + 64] + s1[tid + 128] + s1[tid + 192];
        b = s2[tid] + s2[tid + 64] + s2[tid + 128] + s2[tid + 192];
        atomicAdd(&sum[col], a);
        atomicAdd(&sumsq[col], b);
    }
}

__global__ __launch_bounds__(64)
void bn_finalize_kernel(const float* __restrict__ sum, const float* __restrict__ sumsq,
                        const float* __restrict__ g, const float* __restrict__ beta,
                        float* __restrict__ a_out, float* __restrict__ b_out, int n) {
    int c = threadIdx.x;
    float inv_n = 1.0f / (float)n;
    float mean = sum[c] * inv_n;
    float var  = sumsq[c] * inv_n - mean * mean;
    float inv  = rsqrtf(var + BN_EPS);
    float a = g[c] * inv;
    a_out[c] = a;
    b_out[c] = beta[c] - mean * a;
}

// h_out = relu((agg*in_norm + bias)*a[col] + b2[col] (+ h_in))
__global__ __launch_bounds__(256)
void bn_apply_kernel(const float* __restrict__ agg, const float* __restrict__ innorm,
                     const float* __restrict__ bias,
                     const float* __restrict__ a, const float* __restrict__ b2,
                     const float* __restrict__ hin, float* __restrict__ hout, long count) {
    long i = (long)blockIdx.x * blockDim.x + threadIdx.x;
    if (i >= count) return;
    int c = (int)(i & (H_FEATS - 1));
    int r = (int)(i >> 6);
    float v = agg[i] * innorm[r] + bias[c];
    v = v * a[c] + b2[c];
    if (hin) v += hin[i];
    hout[i] = fmaxf(v, 0.0f);
}

// ---------------- orchestration ----------------

extern "C" void kernel_launch(void* const* d_in, const int* in_sizes, int n_in,
                              void* d_out, int out_size, void* d_ws, size_t ws_size,
                              hipStream_t stream) {
    const int IN_FEATS = 128;
    const float* features = (const float*)d_in[0];
    const int*   src      = (const int*)d_in[1];
    const int*   dst      = (const int*)d_in[2];
    // params flattened in insertion order: (W,b,g,beta) x4, then Wfc, bfc
    const float* Wl[4];
    const float* bl[4];
    const float* gl[4];
    const float* betal[4];
    for (int i = 0; i < 4; ++i) {
        Wl[i]    = (const float*)d_in[3 + 4 * i + 0];
        bl[i]    = (const float*)d_in[3 + 4 * i + 1];
        gl[i]    = (const float*)d_in[3 + 4 * i + 2];
        betal[i] = (const float*)d_in[3 + 4 * i + 3];
    }
    const float* Wfc = (const float*)d_in[19];
    const float* bfc = (const float*)d_in[20];

    const int n_nodes = in_sizes[0] / IN_FEATS;   // 150000
    const int n_edges = in_sizes[1];              // 1200000
    const long N = n_nodes;

    // workspace layout (floats)
    float* ws      = (float*)d_ws;
    float* deg_out = ws;                     // N   -> becomes out_norm
    float* deg_in  = deg_out + N;            // N   -> becomes in_norm
    float* ybuf    = deg_in + N;             // N * 64  (GEMM output / scatter source)
    float* agg     = ybuf + N * 64;          // N * 64  (scatter destination)
    float* hA      = agg + N * 64;           // N * 64
    float* hB      = hA + N * 64;            // N * 64
    float* stats   = hB + N * 64;            // 256: sum[64], sumsq[64], a[64], b2[64]

    // 1. degrees + norms
    zero_kernel<<<512, 256, 0, stream>>>(deg_out, 2 * N);
    degree_kernel<<<(n_edges + 255) / 256, 256, 0, stream>>>(src, dst, deg_out, deg_in, n_edges);
    norm_kernel<<<(int)((2 * N + 255) / 256), 256, 0, stream>>>(deg_out, 2 * N);

    const int rowTiles  = (n_nodes + 15) / 16;
    const int gemmBlks  = (rowTiles + 3) / 4;
    const long cnt64    = N * H_FEATS;

    // 2. GCN layers: y = (x*out_norm)@W  ->  agg = scatter(y)  ->  BN(agg*in_norm + b)
    const float* x = features;
    int K = IN_FEATS;
    float* hin = nullptr;
    float* hout = hA;
    for (int i = 0; i < 4; ++i) {
        size_t shmem = (size_t)K * 64 * sizeof(float);
        gemm_wmma_f32<4><<<gemmBlks, 128, shmem, stream>>>(x, deg_out, Wl[i], nullptr,
                                                           ybuf, n_nodes, K, H_FEATS);
        zero_kernel<<<2048, 256, 0, stream>>>(agg, cnt64);
        scatter_kernel<<<(n_edges + 7) / 8, 256, 0, stream>>>(ybuf, src, dst, agg, n_edges);
        zero_kernel<<<1, 256, 0, stream>>>(stats, 128);
        bn_stats_kernel<<<1024, 256, 0, stream>>>(agg, deg_in, bl[i], stats, stats + 64, n_nodes);
        bn_finalize_kernel<<<1, 64, 0, stream>>>(stats, stats + 64, gl[i], betal[i],
                                                 stats + 128, stats + 192, n_nodes);
        bn_apply_kernel<<<(int)((cnt64 + 255) / 256), 256, 0, stream>>>(
            agg, deg_in, bl[i], stats + 128, stats + 192,
            (i > 0) ? hin : nullptr, hout, cnt64);
        // rotate buffers
        x = hout;
        hin = hout;
        hout = (hout == hA) ? hB : hA;
        K = H_FEATS;
    }

    // 3. final FC: out = h @ Wfc + bfc   (nout = 40 -> 3 col tiles, zero-padded in LDS)
    const int NUM_CLASSES = 40;
    size_t shmem_fc = (size_t)H_FEATS * 48 * sizeof(float);
    gemm_wmma_f32<3><<<gemmBlks, 128, shmem_fc, stream>>>(x, nullptr, Wfc, bfc,
                                                          (float*)d_out, n_nodes,
                                                          H_FEATS, NUM_CLASSES);
}